// CategoryAdder_9019431321845
// MI455X (gfx1250) — compile-verified
//
#include <hip/hip_runtime.h>

// CategoryAdder for MI455X (gfx1250, wave32).
//
// out[b,s,:] = inputs[b,s,:] + (cat!=0 && s!=mask_positions[b] ? table[cat,:] : 0)
//   with cat = categories[b,s]
//
// Memory-bound: ~520 MB true HBM traffic -> ~22-33 us floor @ 23.3 TB/s.
// Strategy:
//   * 2D grid: blockIdx.y = batch b (no integer division on the scalar path),
//     blockIdx.x grid-strides over sequence positions s within the batch
//   * one row (D=512 f32 = 2 KB) per block iteration: 128 lanes x float4
//     -> global_load_b128 / global_store_b128
//   * inputs/out use NT temporal hint (once-through stream; keeps the 10 MB
//     table resident in the 192 MB L2 so the gather is an L2 hit)
//   * categories[row] and mask_pos[b] are block-uniform -> s_load + scalar
//     branch; mask_pos[b] hoisted out of the row loop
//   * global_prefetch_b8 of the next row's input stream (software pipelining)

typedef float f32x4 __attribute__((ext_vector_type(4)));

__global__ __launch_bounds__(128) void category_adder_kernel(
    const float* __restrict__ inputs,          // (B*S, D) f32
    const long long* __restrict__ categories,  // (B*S)    i64
    const long long* __restrict__ mask_pos,    // (B)      i64
    const float* __restrict__ table,           // (NCAT,D) f32
    float* __restrict__ out,                   // (B*S, D) f32
    int S,                                     // sequence length
    int D)                                     // embedding dim (= 4*blockDim.x)
{
    const int t      = threadIdx.x;            // 0 .. D/4-1 (128 for D=512)
    const int b      = blockIdx.y;             // batch index: no div needed
    const int sstep  = gridDim.x;

    // Block-invariant: the masked sequence position for this batch row.
    const long long mp = mask_pos[b];

    const size_t batch_base = (size_t)b * S;   // row offset of this batch

    for (int s = blockIdx.x; s < S; s += sstep) {
        const size_t row = batch_base + s;

        // Software-pipeline the streaming operand: prefetch next iteration's
        // input row (1 byte/lane pulls the full cachelines of the 2 KB row).
        const int ns = s + sstep;
        if (ns < S) {
            __builtin_prefetch(inputs + (batch_base + ns) * D + 4 * t,
                               /*rw=*/0, /*loc=*/1);
        }

        // Block-uniform control data -> s_load + scalar branch.
        const long long cat = categories[row];
        const bool doadd = (cat != 0) && ((long long)s != mp);

        const f32x4* __restrict__ in4  = (const f32x4*)(inputs + row * D) + t;
        f32x4* __restrict__ out4       = (f32x4*)(out + row * D) + t;

        // NT: once-through stream, don't pollute L2 (table lives there).
        f32x4 v = __builtin_nontemporal_load(in4);

        if (doadd) {
            // RT load: table is ~10 MB, fully L2-resident -> cheap gather.
            const f32x4* __restrict__ tb4 =
                (const f32x4*)(table + (size_t)cat * D) + t;
            const f32x4 e = *tb4;
            v.x += e.x; v.y += e.y; v.z += e.z; v.w += e.w;
        }

        __builtin_nontemporal_store(v, out4);
    }
}

extern "C" void kernel_launch(void* const* d_in, const int* in_sizes, int n_in,
                              void* d_out, int out_size, void* d_ws, size_t ws_size,
                              hipStream_t stream) {
    // setup_inputs() order: inputs(f32), categories(i64), mask_positions(i64),
    //                       table(f32)
    const float*     inputs     = (const float*)d_in[0];
    const long long* categories = (const long long*)d_in[1];
    const long long* mask_pos   = (const long long*)d_in[2];
    const float*     table      = (const float*)d_in[3];
    float*           out        = (float*)d_out;

    const int nrows = in_sizes[1];            // B*S
    const int B     = in_sizes[2];            // mask_positions is (B,1)
    const int S     = nrows / B;              // 2048
    const int D     = in_sizes[0] / nrows;    // 512

    const int threads = D / 4;                // 128 lanes = 4 wave32 per block

    // ~16 rows per block iteration pipeline; S/16 x-blocks per batch.
    int gx = (S + 15) / 16;
    if (gx < 1) gx = 1;
    dim3 grid(gx, B, 1);

    category_adder_kernel<<<grid, threads, 0, stream>>>(
        inputs, categories, mask_pos, table, out, S, D);
}